// GCN_38130719654055
// MI455X (gfx1250) — compile-verified
//
#include <hip/hip_runtime.h>
#include <hip/hip_bf16.h>

#define DD 128   // input feature dim
#define HH 128   // hidden dim
#define EPS 1e-5f

typedef float v2f __attribute__((ext_vector_type(2)));
typedef float v8f __attribute__((ext_vector_type(8)));

// ---------------------------------------------------------------------------
// Degree accumulation (in-degree at destinations; self loop added later)
// ---------------------------------------------------------------------------
__global__ void gcn_deg_kernel(const int* __restrict__ dst, float* __restrict__ deg, int E) {
    int e = blockIdx.x * blockDim.x + threadIdx.x;
    if (e < E) atomicAdd(&deg[dst[e]], 1.0f);
}

// dinv[i] = rsqrt(deg[i] + 1)  (self loop => deg+1 > 0 always, so the
// where(deg>0) in the reference is always taken)
__global__ void gcn_dinv_kernel(float* __restrict__ dinv, int N) {
    int i = blockIdx.x * blockDim.x + threadIdx.x;
    if (i < N) dinv[i] = rsqrtf(dinv[i] + 1.0f);
}

// ---------------------------------------------------------------------------
// GEMM: Out[N,128] = A[N,128] @ W[128,128] using V_WMMA_F32_16X16X4_F32.
// 256 threads = 8 waves; block computes a 16-row x 128-col strip; each wave
// owns one 16x16 output tile, accumulating 32 WMMA steps (K=128 / 4).
// ---------------------------------------------------------------------------
__global__ __launch_bounds__(256) void gcn_gemm_wmma_kernel(
    const float* __restrict__ A, const float* __restrict__ W,
    float* __restrict__ Out, int N)
{
    __shared__ float As[16 * DD];  // 8 KB A tile
    const int row0 = blockIdx.x * 16;
    const int tid  = threadIdx.x;

    // Cooperative load of the 16x128 A tile (512 float4, 2 per thread)
    {
        const size_t base = (size_t)row0 * DD;
        float4* Asv = (float4*)As;
        #pragma unroll
        for (int i = 0; i < 2; ++i) {
            int f4  = tid + 256 * i;        // 0..511
            int row = f4 >> 5;              // (f4*4)/128
            float4 v = make_float4(0.f, 0.f, 0.f, 0.f);
            if (row0 + row < N) v = *(const float4*)(A + base + (size_t)f4 * 4);
            Asv[f4] = v;
        }
    }
    __syncthreads();

    const int wave = tid >> 5;        // 0..7  -> column tile
    const int lane = tid & 31;
    const int half = lane >> 4;       // 0: K lo pair, 1: K hi pair
    const int l16  = lane & 15;
    const int col  = wave * 16 + l16; // B/C/D column for this lane
    const int m    = l16;             // A row for this lane

    v8f c = {0.f, 0.f, 0.f, 0.f, 0.f, 0.f, 0.f, 0.f};

    #pragma unroll
    for (int kk = 0; kk < 32; ++kk) {
        const int k0 = kk * 4 + half * 2;
        v2f a, b;
        a.x = As[m * DD + k0];
        a.y = As[m * DD + k0 + 1];
        b.x = W[(size_t)k0 * HH + col];
        b.y = W[(size_t)(k0 + 1) * HH + col];
        // D = A(16x4) * B(4x16) + C  (f32, round-to-nearest-even)
        c = __builtin_amdgcn_wmma_f32_16x16x4_f32(
            /*neg_a=*/false, a, /*neg_b=*/false, b,
            /*c_mod=*/(short)0, c, /*reuse_a=*/false, /*reuse_b=*/false);
    }

    // Store: VGPR r holds M = r + 8*half
    #pragma unroll
    for (int r = 0; r < 8; ++r) {
        int row = row0 + r + half * 8;
        if (row < N) Out[(size_t)row * HH + col] = c[r];
    }
}

// ---------------------------------------------------------------------------
// h[i,c] = bias[c] + dinv[i]^2 * hw[i,c]   (self-loop message + bias init)
// ---------------------------------------------------------------------------
__global__ void gcn_selfloop_bias_kernel(
    const float* __restrict__ hw, const float* __restrict__ dinv,
    const float* __restrict__ bias, float* __restrict__ h, int N)
{
    int idx = blockIdx.x * blockDim.x + threadIdx.x;
    if (idx < N * HH) {
        int i  = idx >> 7;
        int cc = idx & 127;
        float di = dinv[i];
        h[idx] = bias[cc] + di * di * hw[idx];
    }
}

// ---------------------------------------------------------------------------
// Edge scatter: h[dst] += dinv[src]*dinv[dst] * hw[src]; 2 edges per block,
// 128 lanes per edge (one per channel), float atomics to L2.
// ---------------------------------------------------------------------------
__global__ __launch_bounds__(256) void gcn_edge_kernel(
    const int* __restrict__ src, const int* __restrict__ dst,
    const float* __restrict__ dinv, const float* __restrict__ hw,
    float* __restrict__ h, int E)
{
    int e  = blockIdx.x * 2 + (threadIdx.x >> 7);
    int cc = threadIdx.x & 127;
    if (e < E) {
        int s = src[e], d = dst[e];
        float w = dinv[s] * dinv[d];
        atomicAdd(&h[(size_t)d * HH + cc], w * hw[(size_t)s * HH + cc]);
    }
}

// ---------------------------------------------------------------------------
// BatchNorm pass 1: per-channel sum and sum-of-squares into stats[0:128],
// stats[128:256] via LDS partials + global atomics.
// ---------------------------------------------------------------------------
__global__ __launch_bounds__(256) void gcn_bn_stats_kernel(
    const float* __restrict__ h, float* __restrict__ stats, int N)
{
    int cc  = threadIdx.x & 127;
    int sub = threadIdx.x >> 7;   // 0 or 1
    float s = 0.f, sq = 0.f;
    for (int i = blockIdx.x * 2 + sub; i < N; i += gridDim.x * 2) {
        float v = h[(size_t)i * HH + cc];
        s += v; sq += v * v;
    }
    __shared__ float sh[512];
    sh[threadIdx.x]       = s;
    sh[256 + threadIdx.x] = sq;
    __syncthreads();
    if (sub == 0) {
        atomicAdd(&stats[cc],       sh[cc] + sh[128 + cc]);
        atomicAdd(&stats[128 + cc], sh[256 + cc] + sh[256 + 128 + cc]);
    }
}

// BatchNorm pass 2 + ELU (in place)
__global__ void gcn_bn_elu_kernel(
    const float* __restrict__ stats, const float* __restrict__ gamma,
    const float* __restrict__ beta, float* __restrict__ h, int N)
{
    int idx = blockIdx.x * blockDim.x + threadIdx.x;
    if (idx < N * HH) {
        int cc = idx & 127;
        float invN = 1.0f / (float)N;
        float mu   = stats[cc] * invN;
        float var  = stats[128 + cc] * invN - mu * mu;
        float v = gamma[cc] * (h[idx] - mu) * rsqrtf(var + EPS) + beta[cc];
        h[idx] = v > 0.f ? v : expm1f(v);
    }
}

// ---------------------------------------------------------------------------
// Mean pooling per graph
// ---------------------------------------------------------------------------
__global__ void gcn_pool_kernel(
    const float* __restrict__ h, const int* __restrict__ batch,
    float* __restrict__ out, float* __restrict__ counts, int N)
{
    int i  = blockIdx.x;
    int cc = threadIdx.x;      // 128 threads
    if (i < N) {
        int g = batch[i];
        atomicAdd(&out[(size_t)g * HH + cc], h[(size_t)i * HH + cc]);
        if (cc == 0) atomicAdd(&counts[g], 1.0f);
    }
}

__global__ void gcn_pool_div_kernel(float* __restrict__ out,
                                    const float* __restrict__ counts, int G)
{
    int idx = blockIdx.x * blockDim.x + threadIdx.x;
    if (idx < G * HH) {
        float cnt = counts[idx >> 7];
        out[idx] /= fmaxf(cnt, 1.0f);
    }
}

// ---------------------------------------------------------------------------
extern "C" void kernel_launch(void* const* d_in, const int* in_sizes, int n_in,
                              void* d_out, int out_size, void* d_ws, size_t ws_size,
                              hipStream_t stream) {
    const float* x      = (const float*)d_in[0];
    const float* Ws     = (const float*)d_in[1];
    const float* bs     = (const float*)d_in[2];
    const float* gammas = (const float*)d_in[3];
    const float* betas  = (const float*)d_in[4];
    const int*   eidx   = (const int*)d_in[5];
    const int*   batch  = (const int*)d_in[6];
    // d_in[7] = num_graphs (device scalar); G derived from out_size instead.

    const int N = in_sizes[0] / DD;
    const int L = in_sizes[1] / (DD * HH);
    const int E = in_sizes[5] / 2;
    const int G = out_size / HH;

    const int* src = eidx;
    const int* dst = eidx + E;

    // Workspace carve-up
    char* ws = (char*)d_ws;
    float* hw   = (float*)ws;  ws += (size_t)N * HH * sizeof(float);
    float* h    = (float*)ws;  ws += (size_t)N * HH * sizeof(float);
    float* dinv = (float*)ws;  ws += (size_t)N * sizeof(float);
    float* stats= (float*)ws;  ws += 256 * sizeof(float);
    float* cnts = (float*)ws;  ws += (size_t)G * sizeof(float);

    // Degrees -> dinv (in place)
    hipMemsetAsync(dinv, 0, (size_t)N * sizeof(float), stream);
    gcn_deg_kernel<<<(E + 255) / 256, 256, 0, stream>>>(dst, dinv, E);
    gcn_dinv_kernel<<<(N + 255) / 256, 256, 0, stream>>>(dinv, N);

    const float* hin = x;
    for (int l = 0; l < L; ++l) {
        const float* W = Ws + (size_t)l * DD * HH;
        gcn_gemm_wmma_kernel<<<(N + 15) / 16, 256, 0, stream>>>(hin, W, hw, N);
        gcn_selfloop_bias_kernel<<<(N * HH + 255) / 256, 256, 0, stream>>>(
            hw, dinv, bs + (size_t)l * HH, h, N);
        gcn_edge_kernel<<<(E + 1) / 2, 256, 0, stream>>>(src, dst, dinv, hw, h, E);
        hipMemsetAsync(stats, 0, 256 * sizeof(float), stream);
        gcn_bn_stats_kernel<<<512, 256, 0, stream>>>(h, stats, N);
        gcn_bn_elu_kernel<<<(N * HH + 255) / 256, 256, 0, stream>>>(
            stats, gammas + (size_t)l * HH, betas + (size_t)l * HH, h, N);
        hin = h;
    }

    hipMemsetAsync(d_out, 0, (size_t)G * HH * sizeof(float), stream);
    hipMemsetAsync(cnts, 0, (size_t)G * sizeof(float), stream);
    gcn_pool_kernel<<<N, 128, 0, stream>>>(h, batch, (float*)d_out, cnts, N);
    gcn_pool_div_kernel<<<(G * HH + 255) / 256, 256, 0, stream>>>((float*)d_out, cnts, G);
}